// Quantizer_21328807592115
// MI455X (gfx1250) — compile-verified
//
#include <hip/hip_runtime.h>

// Problem constants (match the reference).
constexpr int Bn = 16384;   // rows of x
constexpr int Cn = 4096;    // number of codes
constexpr int Dn = 512;     // feature dim
constexpr int XPITCH = 516; // LDS pitch (mod 64 == 4 -> conflict-free b64 loads)
constexpr int ROWS = 32;    // x-rows per block (2 M-tiles of 16 per wave)

typedef float v2f __attribute__((ext_vector_type(2)));
typedef float v8f __attribute__((ext_vector_type(8)));

// ---------------------------------------------------------------------------
// Kernel 1: c_sq[c] = sum_d codes[c][d]^2.  One wave32 per code.
// ---------------------------------------------------------------------------
__global__ __launch_bounds__(256) void csq_kernel(const float* __restrict__ codes,
                                                  float* __restrict__ csq) {
  const int lane = threadIdx.x & 31;
  const int wave = threadIdx.x >> 5;
  const int c = blockIdx.x * 8 + wave;
  const float4* r4 = (const float4*)(codes + (size_t)c * Dn);
  float s = 0.f;
#pragma unroll
  for (int j = 0; j < Dn / 4 / 32; ++j) {
    float4 v = r4[lane + 32 * j];
    s += v.x * v.x + v.y * v.y + v.z * v.z + v.w * v.w;
  }
#pragma unroll
  for (int m = 16; m; m >>= 1) s += __shfl_xor(s, m);
  if (lane == 0) csq[c] = s;
}

// ---------------------------------------------------------------------------
// Kernel 2: argmin_c ( c_sq[c] - 2 * <x_b, code_c> ) via f32 WMMA.
// Block = 256 threads (8 waves) covers 32 x-rows and ALL 4096 codes.
// Each wave: 2 M-tiles x 2 code chunks -> 4 independent WMMA chains.
// B (codes) fragments are software-pipelined one 16-K window ahead so each
// window's global loads get ~16 WMMAs of latency slack.
// ---------------------------------------------------------------------------
__global__ __launch_bounds__(256) void argmin_kernel(const float* __restrict__ x,
                                                     const float* __restrict__ codes,
                                                     const float* __restrict__ csq,
                                                     int* __restrict__ out_idx) {
  __shared__ float xs[ROWS * XPITCH];  // 66 KB x-tile
  __shared__ float cs[Cn];             // 16 KB c_sq cache
  __shared__ float red_d[8][ROWS];
  __shared__ int   red_i[8][ROWS];

  const int tid = threadIdx.x;
  const int lane = tid & 31;
  const int wave = tid >> 5;
  const int row0 = blockIdx.x * ROWS;

  // Cooperative loads: x tile (float4, coalesced) and c_sq table.
  {
    const float4* xg = (const float4*)(x + (size_t)row0 * Dn);
    for (int i = tid; i < ROWS * (Dn / 4); i += 256) {
      int r = i >> 7;          // row in tile
      int c4 = i & 127;        // float4 column
      float4 v = xg[r * (Dn / 4) + c4];
      *(float4*)&xs[r * XPITCH + c4 * 4] = v;
    }
    const float4* cg = (const float4*)csq;
    float4* cl = (float4*)cs;
    for (int i = tid; i < Cn / 4; i += 256) cl[i] = cg[i];
  }
  __syncthreads();

  const int mrow = lane & 15;   // A: M row / B: N col / D: N col for this lane
  const int half = lane >> 4;   // K sub-pair selector for A/B fragments
  const float* xrow0 = xs + mrow * XPITCH + 2 * half;        // M-tile 0
  const float* xrow1 = xrow0 + 16 * XPITCH;                  // M-tile 1

  float bd[16];
  int   bi[16];
#pragma unroll
  for (int v = 0; v < 16; ++v) { bd[v] = 3.4e38f; bi[v] = 0; }

  for (int q = wave; q < 128; q += 8) {
    const int c0a = 32 * q;        // first 16-code chunk
    const int c0b = 32 * q + 16;   // second 16-code chunk
    const float* ba = codes + (size_t)(c0a + mrow) * Dn + 2 * half;
    const float* bb = codes + (size_t)(c0b + mrow) * Dn + 2 * half;

    // Read the c_sq values before the K loop (no ds stall in the epilogue).
    const float csa = cs[c0a + mrow];
    const float csb = cs[c0b + mrow];
    const int ia = c0a + mrow;
    const int ib = c0b + mrow;

    v8f acc00 = {0.f, 0.f, 0.f, 0.f, 0.f, 0.f, 0.f, 0.f};  // mtile0 x chunkA
    v8f acc10 = {0.f, 0.f, 0.f, 0.f, 0.f, 0.f, 0.f, 0.f};  // mtile1 x chunkA
    v8f acc01 = {0.f, 0.f, 0.f, 0.f, 0.f, 0.f, 0.f, 0.f};  // mtile0 x chunkB
    v8f acc11 = {0.f, 0.f, 0.f, 0.f, 0.f, 0.f, 0.f, 0.f};  // mtile1 x chunkB

    // Prime the B pipeline: window 0 (K = 0..15).
    v2f pb0[4], pb1[4];
#pragma unroll
    for (int j = 0; j < 4; ++j) {
      pb0[j] = *(const v2f*)(ba + 4 * j);
      pb1[j] = *(const v2f*)(bb + 4 * j);
    }

    for (int k0 = 0; k0 < Dn; k0 += 16) {
      // Prefetch next window (wraps on the last iteration; values unused but
      // stay inside the codes row -> no OOB access).
      const int kn = (k0 + 16) & (Dn - 1);
      v2f nb0[4], nb1[4];
#pragma unroll
      for (int j = 0; j < 4; ++j) {
        nb0[j] = *(const v2f*)(ba + kn + 4 * j);
        nb1[j] = *(const v2f*)(bb + kn + 4 * j);
      }

      // Consume the current window: 16 WMMAs over K = k0..k0+15.
#pragma unroll
      for (int j = 0; j < 4; ++j) {
        const int k = k0 + 4 * j;
        v2f a0 = *(const v2f*)(xrow0 + k);  // ds_load (2addr-merged)
        v2f a1 = *(const v2f*)(xrow1 + k);
        acc00 = __builtin_amdgcn_wmma_f32_16x16x4_f32(false, a0, false, pb0[j],
                                                      (short)0, acc00, false, false);
        acc10 = __builtin_amdgcn_wmma_f32_16x16x4_f32(false, a1, false, pb0[j],
                                                      (short)0, acc10, false, false);
        acc01 = __builtin_amdgcn_wmma_f32_16x16x4_f32(false, a0, false, pb1[j],
                                                      (short)0, acc01, false, false);
        acc11 = __builtin_amdgcn_wmma_f32_16x16x4_f32(false, a1, false, pb1[j],
                                                      (short)0, acc11, false, false);
      }

#pragma unroll
      for (int j = 0; j < 4; ++j) { pb0[j] = nb0[j]; pb1[j] = nb1[j]; }
    }

#pragma unroll
    for (int v = 0; v < 8; ++v) {
      // chunk A first (lower indices), then chunk B: strict < keeps first-min.
      float d0a = csa - 2.f * acc00[v];
      if (d0a < bd[v]) { bd[v] = d0a; bi[v] = ia; }
      float d0b = csb - 2.f * acc01[v];
      if (d0b < bd[v]) { bd[v] = d0b; bi[v] = ib; }
      float d1a = csa - 2.f * acc10[v];
      if (d1a < bd[8 + v]) { bd[8 + v] = d1a; bi[8 + v] = ia; }
      float d1b = csb - 2.f * acc11[v];
      if (d1b < bd[8 + v]) { bd[8 + v] = d1b; bi[8 + v] = ib; }
    }
  }

  // Butterfly argmin across the 16 lanes of each half (masks 1,2,4,8 keep
  // halves separate). D layout: slot v of lane l holds row (v + 8*(l>>4)).
#pragma unroll
  for (int mt = 0; mt < 2; ++mt) {
#pragma unroll
    for (int v = 0; v < 8; ++v) {
      float d = bd[mt * 8 + v];
      int   i = bi[mt * 8 + v];
#pragma unroll
      for (int m = 1; m <= 8; m <<= 1) {
        float od = __shfl_xor(d, m);
        int   oi = __shfl_xor(i, m);
        if (od < d || (od == d && oi < i)) { d = od; i = oi; }
      }
      if (mrow == 0) {
        red_d[wave][mt * 16 + v + 8 * half] = d;
        red_i[wave][mt * 16 + v + 8 * half] = i;
      }
    }
  }
  __syncthreads();

  if (tid < ROWS) {
    float d = red_d[0][tid];
    int   i = red_i[0][tid];
#pragma unroll
    for (int w = 1; w < 8; ++w) {
      float od = red_d[w][tid];
      int   oi = red_i[w][tid];
      if (od < d || (od == d && oi < i)) { d = od; i = oi; }
    }
    out_idx[row0 + tid] = i;
  }
}

// ---------------------------------------------------------------------------
// Kernel 3: zero the loss accumulator (runs every call -> deterministic).
// ---------------------------------------------------------------------------
__global__ void zero_kernel(float* loss) { *loss = 0.f; }

// ---------------------------------------------------------------------------
// Kernel 4: gather quantized = codes[idx], write float(idx), accumulate
// loss = (1+BETA) * mean_b ||x_b - q_b||^2  (stop_gradient is identity fwd).
// One block (128 threads) per row; exactly one float4 per thread.
// ---------------------------------------------------------------------------
__global__ __launch_bounds__(128) void gather_kernel(const float* __restrict__ x,
                                                     const float* __restrict__ codes,
                                                     const int* __restrict__ idx,
                                                     float* __restrict__ out_q,
                                                     float* __restrict__ out_idx_f,
                                                     float* __restrict__ loss) {
  __shared__ float warp_s[4];
  const int row = blockIdx.x;
  const int tid = threadIdx.x;
  const int id = idx[row];

  const float4* q4 = (const float4*)(codes + (size_t)id * Dn);
  const float4* x4 = (const float4*)(x + (size_t)row * Dn);
  float4* o4 = (float4*)(out_q + (size_t)row * Dn);

  float4 qv = q4[tid];
  float4 xv = x4[tid];
  o4[tid] = qv;

  float dx = xv.x - qv.x, dy = xv.y - qv.y, dz = xv.z - qv.z, dw = xv.w - qv.w;
  float s = dx * dx + dy * dy + dz * dz + dw * dw;
#pragma unroll
  for (int m = 16; m; m >>= 1) s += __shfl_xor(s, m);

  const int lane = tid & 31, wave = tid >> 5;
  if (lane == 0) warp_s[wave] = s;
  __syncthreads();
  if (tid == 0) {
    float t = warp_s[0] + warp_s[1] + warp_s[2] + warp_s[3];
    atomicAdd(loss, t * (1.25f / (float)Bn));  // (1 + BETA) / B
    out_idx_f[row] = (float)id;
  }
}

// ---------------------------------------------------------------------------
extern "C" void kernel_launch(void* const* d_in, const int* in_sizes, int n_in,
                              void* d_out, int out_size, void* d_ws, size_t ws_size,
                              hipStream_t stream) {
  (void)in_sizes; (void)n_in; (void)out_size; (void)ws_size;
  const float* x = (const float*)d_in[0];       // (B, D) f32
  const float* codes = (const float*)d_in[1];   // (1, C, D) f32

  float* out_q = (float*)d_out;                       // quantized (B*D)
  float* out_idx_f = out_q + (size_t)Bn * Dn;         // indices as float (B)
  float* loss = out_idx_f + Bn;                       // scalar loss

  float* csq = (float*)d_ws;                          // C floats
  int* widx = (int*)(csq + Cn);                       // B ints

  csq_kernel<<<Cn / 8, 256, 0, stream>>>(codes, csq);
  argmin_kernel<<<Bn / ROWS, 256, 0, stream>>>(x, codes, csq, widx);
  zero_kernel<<<1, 1, 0, stream>>>(loss);
  gather_kernel<<<Bn, 128, 0, stream>>>(x, codes, widx, out_q, out_idx_f, loss);
}